// BalSCL_48146583388587
// MI455X (gfx1250) — compile-verified
//
#include <hip/hip_runtime.h>
#include <hip/hip_bf16.h>

#define NUM_CLASSES 100
#define BATCH       4096
#define TWOB        8192
#define FEAT_DIM    128
#define NCOLS       (TWOB + NUM_CLASSES)   // 8292
#define NTILES      ((NCOLS + 15) / 16)    // 519
#define LDS_STRIDE  132                    // 128 + 4 pad -> conflict-free ds_load_b64
#define WAVES_PER_BLOCK 4
#define BLOCK       (WAVES_PER_BLOCK * 32)
#define ROW_TILES   (TWOB / 16)            // 512
#define GRID_MAIN   (ROW_TILES / WAVES_PER_BLOCK)  // 128

typedef __attribute__((ext_vector_type(2))) float v2f;
typedef __attribute__((ext_vector_type(8))) float v8f;

// ---------------------------------------------------------------------------
// Kernel 1: per-class histogram of targets (single block).
// batch_cls_count[c] = 2*hist[c] + 1 (two views + one center).
// ---------------------------------------------------------------------------
__global__ __launch_bounds__(256) void bal_hist_kernel(const int* __restrict__ targets,
                                                       int* __restrict__ hist) {
  __shared__ int h[NUM_CLASSES];
  for (int c = threadIdx.x; c < NUM_CLASSES; c += blockDim.x) h[c] = 0;
  __syncthreads();
  for (int i = threadIdx.x; i < BATCH; i += blockDim.x) atomicAdd(&h[targets[i]], 1);
  __syncthreads();
  for (int c = threadIdx.x; c < NUM_CLASSES; c += blockDim.x) hist[c] = h[c];
}

// ---------------------------------------------------------------------------
// Kernel 2: fused GEMM + online weighted log-sum-exp + positive accumulation.
// Each wave computes a 16-row tile of the 8192 x 8292 logits matrix via
// V_WMMA_F32_16X16X4_F32 (two independent 16-deep accumulator chains, K=128),
// never materializing the logits. Column tiles are staged into LDS with
// GLOBAL_LOAD_ASYNC_TO_LDS_B128 (ASYNCcnt) — no VGPR round-trip.
// row_loss[i] = m_i + log(S_i) - P_i/(N_i + 1e-9)   (loss = mean over rows)
// ---------------------------------------------------------------------------
__global__ __launch_bounds__(BLOCK) void bal_main_kernel(const float* __restrict__ centers,
                                                         const float* __restrict__ feats,
                                                         const int*   __restrict__ targets,
                                                         const int*   __restrict__ hist,
                                                         float*       __restrict__ row_loss) {
  __shared__ float ldsB[16 * LDS_STRIDE];
  __shared__ int   ldsHist[NUM_CLASSES];

  const int tid  = threadIdx.x;
  const int wave = tid >> 5;
  const int lane = tid & 31;
  const int hi   = lane >> 4;   // half-wave: 0 or 1
  const int ln   = lane & 15;   // position within half

  for (int c = tid; c < NUM_CLASSES; c += BLOCK) ldsHist[c] = hist[c];

  const int rowTile = blockIdx.x * WAVES_PER_BLOCK + wave;  // 0..511
  const int rowBase = rowTile * 16;
  const float invT  = 10.0f;  // 1 / TEMPERATURE, folded into A

  // A fragments for K=128: chunk k supplies K = 4k + 2*hi + {0,1}, M = ln.
  v2f a[32];
  {
    const float* fr = feats + (size_t)(rowBase + ln) * FEAT_DIM;
    #pragma unroll
    for (int k = 0; k < 32; ++k) {
      a[k].x = fr[4 * k + 2 * hi + 0] * invT;
      a[k].y = fr[4 * k + 2 * hi + 1] * invT;
    }
  }

  // Targets for the 8 rows this lane accumulates (rows 8*hi + r).
  int rt[8];
  #pragma unroll
  for (int r = 0; r < 8; ++r) rt[r] = targets[(rowBase + 8 * hi + r) & (BATCH - 1)];

  // Online stats per owned row: running max, weighted exp-sum, positive logit
  // sum, positive count (each lane covers columns j == (colTile*16 + ln)).
  float m[8], S[8], P[8], N[8];
  #pragma unroll
  for (int r = 0; r < 8; ++r) { m[r] = -1e30f; S[r] = 0.0f; P[r] = 0.0f; N[r] = 0.0f; }

  for (int t = 0; t < NTILES; ++t) {
    const int colBase = t * 16;

    __syncthreads();   // previous tile fully consumed before overwrite

    // Async staging: 16 columns x 128 floats (512 x b128), direct to LDS,
    // tracked by ASYNCcnt. Padded columns clamp the source address; their
    // (garbage) logits stay confined to columns the epilogue skips.
    #pragma unroll
    for (int it = 0; it < (16 * 32) / BLOCK; ++it) {
      const int q    = tid + it * BLOCK;
      const int col  = q >> 5;          // 32 float4 per column
      const int off4 = q & 31;
      int j = colBase + col;
      j = (j < NCOLS) ? j : (NCOLS - 1);
      const float* src = (j < TWOB) ? feats + (size_t)j * FEAT_DIM
                                    : centers + (size_t)(j - TWOB) * FEAT_DIM;
      src += off4 * 4;
      const unsigned ldsOff =
          (unsigned)(uintptr_t)&ldsB[col * LDS_STRIDE + off4 * 4];
      asm volatile("global_load_async_to_lds_b128 %0, %1, off"
                   :: "v"(ldsOff), "v"(src) : "memory");
    }
    // Prefetch next tile into near caches while we compute on this one.
    if (t + 1 < NTILES) {
      const int jn = (t + 1) * 16 + ln;
      if (jn < NCOLS) {
        const float* p = (jn < TWOB) ? feats + (size_t)jn * FEAT_DIM
                                     : centers + (size_t)(jn - TWOB) * FEAT_DIM;
        __builtin_prefetch(p, 0, 3);
      }
    }
#if __has_builtin(__builtin_amdgcn_s_wait_asynccnt)
    __builtin_amdgcn_s_wait_asynccnt(0);
#else
    asm volatile("s_wait_asynccnt 0x0" ::: "memory");
#endif
    __syncthreads();

    // 16x16 logits tile: D = A x B, K=128 via two independent WMMA chains.
    v8f acc0 = {};
    v8f acc1 = {};
    #pragma unroll
    for (int k = 0; k < 16; ++k) {
      v2f b0 = *(const v2f*)&ldsB[ln * LDS_STRIDE + 8 * k + 0 + 2 * hi];
      v2f b1 = *(const v2f*)&ldsB[ln * LDS_STRIDE + 8 * k + 4 + 2 * hi];
      acc0 = __builtin_amdgcn_wmma_f32_16x16x4_f32(
          false, a[2 * k + 0], false, b0, (short)0, acc0, false, false);
      acc1 = __builtin_amdgcn_wmma_f32_16x16x4_f32(
          false, a[2 * k + 1], false, b1, (short)0, acc1, false, false);
    }
    const v8f acc = acc0 + acc1;

    // Per-lane epilogue: column j = colBase + ln, rows 8*hi + r (r = vgpr idx).
    const int  j      = colBase + ln;
    const bool validj = (j < NCOLS);
    int tj = 0; float cj = 1.0f;
    if (validj) {
      tj = (j < TWOB) ? targets[j & (BATCH - 1)] : (j - TWOB);
      cj = (float)(2 * ldsHist[tj] + 1);   // batch_cls_count of column's class
    }
    #pragma unroll
    for (int r = 0; r < 8; ++r) {
      const int i = rowBase + 8 * hi + r;
      if (validj && j != i) {
        const float v   = acc[r];           // logit (already / T)
        const bool  pos = (tj == rt[r]);
        const float w   = cj - (pos ? 1.0f : 0.0f);
        if (v > m[r]) {                     // rescale running sum
          S[r] = S[r] * __expf(m[r] - v) + 1.0f / w;
          m[r] = v;
        } else {
          S[r] += __expf(v - m[r]) / w;
        }
        if (pos) { P[r] += v; N[r] += 1.0f; }
      }
    }
  }

  // Butterfly merge across the 16 lanes that share each row.
  #pragma unroll
  for (int off = 1; off < 16; off <<= 1) {
    #pragma unroll
    for (int r = 0; r < 8; ++r) {
      const float mo = __shfl_xor(m[r], off, 32);
      const float So = __shfl_xor(S[r], off, 32);
      const float Po = __shfl_xor(P[r], off, 32);
      const float No = __shfl_xor(N[r], off, 32);
      const float mn = fmaxf(m[r], mo);
      S[r] = S[r] * __expf(m[r] - mn) + So * __expf(mo - mn);
      m[r] = mn;
      P[r] += Po;
      N[r] += No;
    }
  }

  // Lanes ln<8 publish the 16 row results of this wave's tile.
  #pragma unroll
  for (int r = 0; r < 8; ++r) {
    if (ln == r) {
      const int i = rowBase + 8 * hi + r;
      row_loss[i] = m[r] + logf(S[r]) - P[r] / (N[r] + 1e-9f);
    }
  }
}

// ---------------------------------------------------------------------------
// Kernel 3: mean over 8192 per-row losses (single block).
// ---------------------------------------------------------------------------
__global__ __launch_bounds__(256) void bal_reduce_kernel(const float* __restrict__ row_loss,
                                                         float* __restrict__ out) {
  __shared__ float sm[256];
  float s = 0.0f;
  for (int i = threadIdx.x; i < TWOB; i += 256) s += row_loss[i];
  sm[threadIdx.x] = s;
  __syncthreads();
  for (int w = 128; w > 0; w >>= 1) {
    if (threadIdx.x < (unsigned)w) sm[threadIdx.x] += sm[threadIdx.x + w];
    __syncthreads();
  }
  if (threadIdx.x == 0) out[0] = sm[0] / (float)TWOB;
}

// ---------------------------------------------------------------------------
extern "C" void kernel_launch(void* const* d_in, const int* in_sizes, int n_in,
                              void* d_out, int out_size, void* d_ws, size_t ws_size,
                              hipStream_t stream) {
  const float* centers = (const float*)d_in[0];   // [100, 128]  f32
  const float* feats   = (const float*)d_in[1];   // [8192, 128] f32
  const int*   targets = (const int*)d_in[2];     // [4096]      i32

  int*   hist     = (int*)d_ws;                         // 128 ints (100 used)
  float* row_loss = (float*)((char*)d_ws + 512);        // 8192 floats

  bal_hist_kernel<<<1, 256, 0, stream>>>(targets, hist);
  bal_main_kernel<<<GRID_MAIN, BLOCK, 0, stream>>>(centers, feats, targets, hist, row_loss);
  bal_reduce_kernel<<<1, 256, 0, stream>>>(row_loss, (float*)d_out);
}